// MessagePassing_9775345566164
// MI455X (gfx1250) — compile-verified
//
#include <hip/hip_runtime.h>

// Graph message passing on MI455X (gfx1250, wave32).
// d_in[0]: nodes  f32 [100000*128]
// d_in[1]: edges  f32 [1600000*128]   (passthrough)
// d_in[2]: adjacency i32 [1600000*2]
// d_out  : new_nodes (12.8M f32) ++ edges (204.8M f32)
// d_ws   : deg f32 [100000]

#define N_NODES 100000
#define N_EDGES 1600000
#define D_FEAT  128

typedef float vf4 __attribute__((ext_vector_type(4)));
typedef int   vi2 __attribute__((ext_vector_type(2)));

// ---------------------------------------------------------------------------
// Kernel 1: accumulator init. d_out[node region] = nodes, deg = 0.
// Regular-temporal loads/stores: this 51MB table must stay L2-resident for
// the scatter phase's gathers + atomics.
// ---------------------------------------------------------------------------
__global__ __launch_bounds__(256) void mp_init(const vf4* __restrict__ nodes4,
                                               vf4* __restrict__ out4,
                                               float* __restrict__ deg) {
  const int i = blockIdx.x * 256 + threadIdx.x;   // exactly N_NODES*D_FEAT/4 threads
  out4[i] = nodes4[i];
  if (i < N_NODES) deg[i] = 0.0f;
}

// ---------------------------------------------------------------------------
// Kernel 2: edge scatter. One wave32 per edge: 32 lanes x float4 = 128 floats.
// Gathers are 512B coalesced rows from the L2-resident node table; scatters
// are fire-and-forget global_atomic_add_f32 (no-return -> STOREcnt path).
// ---------------------------------------------------------------------------
__global__ __launch_bounds__(256) void mp_scatter(const float* __restrict__ nodes,
                                                  const vi2* __restrict__ adj,
                                                  float* __restrict__ acc,
                                                  float* __restrict__ deg) {
  const int lane = threadIdx.x & 31;
  const int e = blockIdx.x * 8 + (threadIdx.x >> 5);  // N_EDGES % 8 == 0
  const vi2 sd = adj[e];
  const int s = sd.x;
  const int d = sd.y;
  const vf4 vd = ((const vf4*)(nodes + (size_t)d * D_FEAT))[lane];
  const vf4 vs = ((const vf4*)(nodes + (size_t)s * D_FEAT))[lane];
  float* os = acc + (size_t)s * D_FEAT + lane * 4;
  float* od = acc + (size_t)d * D_FEAT + lane * 4;
  atomicAdd(os + 0, vd.x); atomicAdd(os + 1, vd.y);
  atomicAdd(os + 2, vd.z); atomicAdd(os + 3, vd.w);
  atomicAdd(od + 0, vs.x); atomicAdd(od + 1, vs.y);
  atomicAdd(od + 2, vs.z); atomicAdd(od + 3, vs.w);
  if (lane == 0) {
    atomicAdd(deg + s, 1.0f);
    atomicAdd(deg + d, 1.0f);
  }
}

// ---------------------------------------------------------------------------
// Kernel 3: normalize in place: out = (nodes + agg) / (deg + 1).
// Final write is non-temporal (never re-read; don't pollute L2).
// ---------------------------------------------------------------------------
__global__ __launch_bounds__(256) void mp_normalize(vf4* __restrict__ out4,
                                                    const float* __restrict__ deg) {
  const int i = blockIdx.x * 256 + threadIdx.x;   // exactly N_NODES*32 threads
  const int node = i >> 5;
  const float denom = deg[node] + 1.0f;
  vf4 v = out4[i];
  v = v / denom;
  __builtin_nontemporal_store(v, &out4[i]);
}

// ---------------------------------------------------------------------------
// Kernel 4: edge passthrough copy (0.82 GB each way, pure streaming).
// Preferred path: gfx1250 async global<->LDS DMA (ASYNCcnt), which keeps the
// data out of VGPRs entirely. Fallback: non-temporal B128 copies.
// Grid is sized so total4 % (blocks*256) == 0 -> EXEC is all-ones for every
// async issue, per ISA guidance.
// ---------------------------------------------------------------------------
#define EDGE_COPY_BLOCKS 10000   // 10000*256 threads, 20 iterations exactly

#if __has_builtin(__builtin_amdgcn_global_load_async_to_lds_b128) && \
    __has_builtin(__builtin_amdgcn_global_store_async_from_lds_b128) && \
    __has_builtin(__builtin_amdgcn_s_wait_asynccnt)
#define MP_HAVE_ASYNC 1

// Builtin param types per hipcc diagnostic:
//   arg0: int __attribute__((vector_size(16))) __device__(AS1) *
//   arg1: LDS-side pointer, same pointee (AS3)
//   arg2: i32 offset, arg3: i32 cpol
typedef int v4i __attribute__((vector_size(16)));
typedef __attribute__((address_space(1))) v4i glb_v4i;
typedef __attribute__((address_space(3))) v4i lds_v4i;

__global__ __launch_bounds__(256) void mp_edge_copy_async(const vf4* __restrict__ src,
                                                          vf4* __restrict__ dst,
                                                          unsigned total4) {
  __shared__ vf4 stage[256];
  const unsigned stride = gridDim.x * 256;
  // AS(3) pointer from generic: low 32 bits of a flat LDS address are the LDS offset.
  lds_v4i* lp = (lds_v4i*)(unsigned)(unsigned long long)&stage[threadIdx.x];
  for (unsigned i = blockIdx.x * 256 + threadIdx.x; i < total4; i += stride) {
    glb_v4i* gs = (glb_v4i*)(unsigned long long)(src + i);
    glb_v4i* gd = (glb_v4i*)(unsigned long long)(dst + i);
    __builtin_amdgcn_global_load_async_to_lds_b128(gs, lp, 0, 0);
    __builtin_amdgcn_s_wait_asynccnt(0);
    __builtin_amdgcn_global_store_async_from_lds_b128(gd, lp, 0, 0);
    __builtin_amdgcn_s_wait_asynccnt(0);
  }
}
#endif

__global__ __launch_bounds__(256) void mp_edge_copy(const vf4* __restrict__ src,
                                                    vf4* __restrict__ dst,
                                                    unsigned total4) {
  const unsigned stride = gridDim.x * 256;
  for (unsigned i = blockIdx.x * 256 + threadIdx.x; i < total4; i += stride) {
    vf4 v = __builtin_nontemporal_load(&src[i]);
    __builtin_nontemporal_store(v, &dst[i]);
  }
}

// ---------------------------------------------------------------------------
extern "C" void kernel_launch(void* const* d_in, const int* in_sizes, int n_in,
                              void* d_out, int out_size, void* d_ws, size_t ws_size,
                              hipStream_t stream) {
  const float* nodes = (const float*)d_in[0];
  const float* edges = (const float*)d_in[1];
  const int*   adj   = (const int*)d_in[2];

  float* out_nodes = (float*)d_out;
  float* out_edges = out_nodes + (size_t)N_NODES * D_FEAT;
  float* deg       = (float*)d_ws;

  (void)in_sizes; (void)n_in; (void)out_size; (void)ws_size;

  const int node4 = N_NODES * D_FEAT / 4;          // 3,200,000 (divisible by 256)
  mp_init<<<node4 / 256, 256, 0, stream>>>((const vf4*)nodes, (vf4*)out_nodes, deg);

  mp_scatter<<<N_EDGES / 8, 256, 0, stream>>>(nodes, (const vi2*)adj, out_nodes, deg);

  mp_normalize<<<(N_NODES * 32) / 256, 256, 0, stream>>>((vf4*)out_nodes, deg);

  const unsigned total4 = (unsigned)((size_t)N_EDGES * D_FEAT / 4);  // 51,200,000
#ifdef MP_HAVE_ASYNC
  mp_edge_copy_async<<<EDGE_COPY_BLOCKS, 256, 0, stream>>>((const vf4*)edges,
                                                           (vf4*)out_edges, total4);
#else
  mp_edge_copy<<<EDGE_COPY_BLOCKS, 256, 0, stream>>>((const vf4*)edges,
                                                     (vf4*)out_edges, total4);
#endif
}